// CGNN_46377056862932
// MI455X (gfx1250) — compile-verified
//
#include <hip/hip_runtime.h>
#include <math.h>

// ---------------------------------------------------------------------------
// CGNN layer on gfx1250 (MI455X).  fp32 WMMA (V_WMMA_F32_16X16X4_F32) GEMMs
// with fused epilogue reductions (attention gate + classifier collapse into
// the GEMM kernels via shfl + LDS ds_add_f32), plus an L2-resident
// gather/scatter edge phase.  Per-edge attention is factored to per-node math
// (alpha depends only on the source node), removing the 850k-edge 128x128
// GEMM entirely.
// ---------------------------------------------------------------------------

typedef float v2f __attribute__((ext_vector_type(2)));
typedef float v8f __attribute__((ext_vector_type(8)));

#define IN_DIM 256
#define HID    128
#define HALF   64

__device__ __forceinline__ v8f wmma_f32(v2f a, v2f b, v8f c) {
  // D = A(16x4 f32) * B(4x16 f32) + C(16x16 f32)
  return __builtin_amdgcn_wmma_f32_16x16x4_f32(
      /*neg_a=*/false, a, /*neg_b=*/false, b,
      /*c_mod=*/(short)0, c, /*reuse_a=*/false, /*reuse_b=*/false);
}

__device__ __forceinline__ float lrelu(float v) { return v > 0.f ? v : 0.01f * v; }
__device__ __forceinline__ float sigmoidf(float v) { return 1.0f / (1.0f + __expf(-v)); }

// ---------------- GEMM 1: h = leaky_relu(x @ W_in + b_in) ------------------
__global__ __launch_bounds__(256)
void k_gemm_in(const float* __restrict__ x, const float* __restrict__ W,
               const float* __restrict__ bias, float* __restrict__ h, int n) {
  const int lane  = threadIdx.x & 31;
  const int wid   = threadIdx.x >> 5;          // 8 waves -> 8 N-tiles (HID=128)
  const int mbase = blockIdx.x * 16;
  const int nb    = wid * 16;
  const int m     = lane & 15;                 // A-row (per lane half) == B-col
  const int koff  = (lane < 16) ? 0 : 2;       // K-pair split across lane halves
  int arow = mbase + m; if (arow >= n) arow = n - 1;
  const float* ap = x + (size_t)arow * IN_DIM + koff;
  const float* bp = W + (size_t)koff * HID + nb + m;
  v8f c = {0.f,0.f,0.f,0.f,0.f,0.f,0.f,0.f};
  #pragma unroll 4
  for (int k = 0; k < IN_DIM; k += 4) {
    v2f a; a.x = ap[0];  a.y = ap[1];
    v2f b; b.x = bp[0];  b.y = bp[HID];
    c = wmma_f32(a, b, c);
    ap += 4; bp += 4 * HID;
  }
  const float bv    = bias[nb + m];
  const int   rbase = mbase + ((lane & 16) ? 8 : 0);
  float* hp = h + (size_t)rbase * HID + nb + m;
  if (mbase + 15 < n) {                         // full tile: branch-free stores
    #pragma unroll
    for (int r = 0; r < 8; ++r) hp[(size_t)r * HID] = lrelu(c[r] + bv);
  } else {
    #pragma unroll
    for (int r = 0; r < 8; ++r)
      if (rbase + r < n) hp[(size_t)r * HID] = lrelu(c[r] + bv);
  }
}

// ------- GEMM 2: xn = h[:, :64]@W_nor + b ; xa = h[:, 64:]@W_abnor + b -----
__global__ __launch_bounds__(256)
void k_gemm_split(const float* __restrict__ h,
                  const float* __restrict__ Wn, const float* __restrict__ bn,
                  const float* __restrict__ Wa, const float* __restrict__ ba,
                  float* __restrict__ xn, float* __restrict__ xa, int n) {
  const int lane  = threadIdx.x & 31;
  const int wid   = threadIdx.x >> 5;
  const int mbase = blockIdx.x * 16;
  const int nb    = wid * 16;
  const int m     = lane & 15;
  const int koff  = (lane < 16) ? 0 : 2;
  int arow = mbase + m; if (arow >= n) arow = n - 1;
  const float* apn = h + (size_t)arow * HID + koff;          // first half
  const float* apa = h + (size_t)arow * HID + HALF + koff;   // second half
  const float* bpn = Wn + (size_t)koff * HID + nb + m;
  const float* bpa = Wa + (size_t)koff * HID + nb + m;
  v8f c1 = {0.f,0.f,0.f,0.f,0.f,0.f,0.f,0.f};
  v8f c2 = {0.f,0.f,0.f,0.f,0.f,0.f,0.f,0.f};
  #pragma unroll 4
  for (int k = 0; k < HALF; k += 4) {
    v2f a1; a1.x = apn[0]; a1.y = apn[1];
    v2f b1; b1.x = bpn[0]; b1.y = bpn[HID];
    c1 = wmma_f32(a1, b1, c1);
    v2f a2; a2.x = apa[0]; a2.y = apa[1];
    v2f b2; b2.x = bpa[0]; b2.y = bpa[HID];
    c2 = wmma_f32(a2, b2, c2);
    apn += 4; apa += 4; bpn += 4 * HID; bpa += 4 * HID;
  }
  const float bvn = bn[nb + m], bva = ba[nb + m];
  const int rbase = mbase + ((lane & 16) ? 8 : 0);
  const size_t o0 = (size_t)rbase * HID + nb + m;
  if (mbase + 15 < n) {
    #pragma unroll
    for (int r = 0; r < 8; ++r) {
      xn[o0 + (size_t)r * HID] = c1[r] + bvn;
      xa[o0 + (size_t)r * HID] = c2[r] + bva;
    }
  } else {
    #pragma unroll
    for (int r = 0; r < 8; ++r) {
      if (rbase + r < n) {
        xn[o0 + (size_t)r * HID] = c1[r] + bvn;
        xa[o0 + (size_t)r * HID] = c2[r] + bva;
      }
    }
  }
}

// -- GEMM 3 + gate (fused): alpha = sigmoid(tanh((xn+xa)@W_att+b_att)@v_att) -
// Per-row dot with v_att reduced in-register (shfl over 16 lanes) then across
// the block's 8 waves via LDS float atomics.  No a_att buffer is ever stored.
__global__ __launch_bounds__(256)
void k_gemm_att_alpha(const float* __restrict__ xn, const float* __restrict__ xa,
                      const float* __restrict__ W, const float* __restrict__ bias,
                      const float* __restrict__ v_att, float* __restrict__ alpha,
                      int n) {
  __shared__ float red[16];
  const int lane  = threadIdx.x & 31;
  const int wid   = threadIdx.x >> 5;
  const int mbase = blockIdx.x * 16;
  const int nb    = wid * 16;
  const int m     = lane & 15;
  const int koff  = (lane < 16) ? 0 : 2;
  if (threadIdx.x < 16) red[threadIdx.x] = 0.0f;
  __syncthreads();
  int arow = mbase + m; if (arow >= n) arow = n - 1;
  const float* pn = xn + (size_t)arow * HID + koff;
  const float* pa = xa + (size_t)arow * HID + koff;
  const float* bp = W + (size_t)koff * HID + nb + m;
  v8f c = {0.f,0.f,0.f,0.f,0.f,0.f,0.f,0.f};
  #pragma unroll 4
  for (int k = 0; k < HID; k += 4) {
    v2f a; a.x = pn[0] + pa[0];  a.y = pn[1] + pa[1];
    v2f b; b.x = bp[0];          b.y = bp[HID];
    c = wmma_f32(a, b, c);
    pn += 4; pa += 4; bp += 4 * HID;
  }
  const float bv   = bias[nb + m];
  const float vcol = v_att[nb + m];
  float pr[8];
  #pragma unroll
  for (int r = 0; r < 8; ++r) pr[r] = tanhf(c[r] + bv) * vcol;
  #pragma unroll
  for (int off = 8; off; off >>= 1) {          // sum over 16 cols in lane half
    #pragma unroll
    for (int r = 0; r < 8; ++r) pr[r] += __shfl_xor(pr[r], off, 32);
  }
  if ((lane & 15) == 0) {                      // lane 0 -> rows 0-7, lane 16 -> 8-15
    const int rb = (lane >> 4) * 8;
    #pragma unroll
    for (int r = 0; r < 8; ++r) atomicAdd(&red[rb + r], pr[r]);
  }
  __syncthreads();
  if (threadIdx.x < 16) {
    const int row = mbase + threadIdx.x;
    if (row < n) alpha[row] = sigmoidf(red[threadIdx.x]);
  }
}

// - GEMM 4 + classifier (fused): out = lrelu((dinv*aggr)@W_upd+b_upd)@W_cls+b -
__global__ __launch_bounds__(256)
void k_gemm_upd_cls(const float* __restrict__ aggr, const float* __restrict__ dinv,
                    const float* __restrict__ W, const float* __restrict__ bias,
                    const float* __restrict__ Wc, const float* __restrict__ bc,
                    float* __restrict__ out, int n) {
  __shared__ float red0[16];
  __shared__ float red1[16];
  const int lane  = threadIdx.x & 31;
  const int wid   = threadIdx.x >> 5;
  const int mbase = blockIdx.x * 16;
  const int nb    = wid * 16;
  const int m     = lane & 15;
  const int koff  = (lane < 16) ? 0 : 2;
  if (threadIdx.x < 16) { red0[threadIdx.x] = 0.0f; red1[threadIdx.x] = 0.0f; }
  __syncthreads();
  int arow = mbase + m; if (arow >= n) arow = n - 1;
  const float scale = dinv[arow];
  const float* ap = aggr + (size_t)arow * HID + koff;
  const float* bp = W + (size_t)koff * HID + nb + m;
  v8f c = {0.f,0.f,0.f,0.f,0.f,0.f,0.f,0.f};
  #pragma unroll 4
  for (int k = 0; k < HID; k += 4) {
    v2f a; a.x = ap[0] * scale;  a.y = ap[1] * scale;
    v2f b; b.x = bp[0];          b.y = bp[HID];
    c = wmma_f32(a, b, c);
    ap += 4; bp += 4 * HID;
  }
  const float bv  = bias[nb + m];
  const float vc0 = Wc[(nb + m) * 2 + 0];
  const float vc1 = Wc[(nb + m) * 2 + 1];
  float p0[8], p1[8];
  #pragma unroll
  for (int r = 0; r < 8; ++r) {
    float v = lrelu(c[r] + bv);
    p0[r] = v * vc0;
    p1[r] = v * vc1;
  }
  #pragma unroll
  for (int off = 8; off; off >>= 1) {
    #pragma unroll
    for (int r = 0; r < 8; ++r) {
      p0[r] += __shfl_xor(p0[r], off, 32);
      p1[r] += __shfl_xor(p1[r], off, 32);
    }
  }
  if ((lane & 15) == 0) {
    const int rb = (lane >> 4) * 8;
    #pragma unroll
    for (int r = 0; r < 8; ++r) {
      atomicAdd(&red0[rb + r], p0[r]);
      atomicAdd(&red1[rb + r], p1[r]);
    }
  }
  __syncthreads();
  if (threadIdx.x < 16) {
    const int row = mbase + threadIdx.x;
    if (row < n) {
      out[(size_t)row * 2 + 0] = red0[threadIdx.x] + bc[0];
      out[(size_t)row * 2 + 1] = red1[threadIdx.x] + bc[1];
    }
  }
}

// ------------------------------ small kernels ------------------------------
__global__ void k_fill(float* __restrict__ p, float v, int n) {
  int i = blockIdx.x * blockDim.x + threadIdx.x;
  if (i < n) p[i] = v;
}

__global__ void k_deg_acc(const int* __restrict__ col, float* __restrict__ deg, int e) {
  int i = blockIdx.x * blockDim.x + threadIdx.x;
  if (i < e) atomicAdd(&deg[col[i]], 1.0f);
}

__global__ void k_dinv(const float* __restrict__ deg, float* __restrict__ dinv, int n) {
  int i = blockIdx.x * blockDim.x + threadIdx.x;
  if (i < n) dinv[i] = rsqrtf(deg[i]);   // deg >= 1 always (self loop)
}

// dm[i,:] = dinv[i] * (alpha[i]*xn[i,:] + (1-alpha[i])*xa[i,:])
__global__ void k_dm(const float* __restrict__ xn, const float* __restrict__ xa,
                     const float* __restrict__ alpha, const float* __restrict__ dinv,
                     float* __restrict__ dm, int n) {
  int t = blockIdx.x * blockDim.x + threadIdx.x;
  int i = t >> 5, lane = t & 31;
  if (i >= n) return;
  float al = alpha[i], di = dinv[i];
  size_t idx = (size_t)i * HID + lane * 4;
  float4 n4 = *(const float4*)(xn + idx);
  float4 a4 = *(const float4*)(xa + idx);
  float4 r;
  r.x = di * (al * n4.x + (1.f - al) * a4.x);
  r.y = di * (al * n4.y + (1.f - al) * a4.y);
  r.z = di * (al * n4.z + (1.f - al) * a4.z);
  r.w = di * (al * n4.w + (1.f - al) * a4.w);
  *(float4*)(dm + idx) = r;
}

// aggr[col,:] += dinv[col] * dm[row,:]  (one wave per edge; self loops appended)
__global__ __launch_bounds__(256)
void k_scatter(const int* __restrict__ row, const int* __restrict__ col,
               const float* __restrict__ dm, const float* __restrict__ dinv,
               float* __restrict__ aggr, int e, int n) {
  const int lane = threadIdx.x & 31;
  const int wid  = threadIdx.x >> 5;
  long long eidx = (long long)blockIdx.x * 8 + wid;
  if (eidx >= (long long)e + n) return;
  int r, c;
  if (eidx < e) { r = row[eidx]; c = col[eidx]; }
  else          { r = c = (int)(eidx - e); }
  float di = dinv[c];
  float4 v = *(const float4*)(dm + (size_t)r * HID + lane * 4);
  float* dst = aggr + (size_t)c * HID + lane * 4;
  atomicAdd(dst + 0, di * v.x);
  atomicAdd(dst + 1, di * v.y);
  atomicAdd(dst + 2, di * v.z);
  atomicAdd(dst + 3, di * v.w);
}

// ---------------------------------------------------------------------------
extern "C" void kernel_launch(void* const* d_in, const int* in_sizes, int n_in,
                              void* d_out, int out_size, void* d_ws, size_t ws_size,
                              hipStream_t stream) {
  const float* x     = (const float*)d_in[0];
  const int*   ei    = (const int*)d_in[1];
  const float* W_in  = (const float*)d_in[2];
  const float* b_in  = (const float*)d_in[3];
  const float* W_nor = (const float*)d_in[4];
  const float* b_nor = (const float*)d_in[5];
  const float* W_ab  = (const float*)d_in[6];
  const float* b_ab  = (const float*)d_in[7];
  const float* W_att = (const float*)d_in[8];
  const float* b_att = (const float*)d_in[9];
  const float* v_att = (const float*)d_in[10];
  const float* W_upd = (const float*)d_in[11];
  const float* b_upd = (const float*)d_in[12];
  const float* W_cls = (const float*)d_in[13];
  const float* b_cls = (const float*)d_in[14];

  const int n = in_sizes[0] / IN_DIM;     // 50000
  const int e = in_sizes[1] / 2;          // 800000
  const int* row = ei;                    // edge_index[0]
  const int* col = ei + e;                // edge_index[1]

  const size_t nh = (size_t)n * HID;
  float* buf0  = (float*)d_ws;            // h      -> dm
  float* buf1  = buf0 + nh;               // xn
  float* buf2  = buf1 + nh;               // xa
  float* buf3  = buf2 + nh;               // aggr
  float* deg   = buf3 + nh;
  float* dinv  = deg + n;
  float* alpha = dinv + n;

  const int mt = (n + 15) / 16;           // 3125 M-tiles

  // degree / normalization + aggr zeroing (independent of GEMM chain)
  k_fill<<<(n + 255) / 256, 256, 0, stream>>>(deg, 1.0f, n);
  k_deg_acc<<<(e + 255) / 256, 256, 0, stream>>>(col, deg, e);
  k_dinv<<<(n + 255) / 256, 256, 0, stream>>>(deg, dinv, n);
  k_fill<<<((int)nh + 255) / 256, 256, 0, stream>>>(buf3, 0.0f, (int)nh);

  // node-side GEMM chain (fp32 WMMA, fused epilogues)
  k_gemm_in<<<mt, 256, 0, stream>>>(x, W_in, b_in, buf0, n);
  k_gemm_split<<<mt, 256, 0, stream>>>(buf0, W_nor, b_nor, W_ab, b_ab, buf1, buf2, n);
  k_gemm_att_alpha<<<mt, 256, 0, stream>>>(buf1, buf2, W_att, b_att, v_att, alpha, n);
  k_dm<<<((int)((size_t)n * 32) + 255) / 256, 256, 0, stream>>>(buf1, buf2, alpha, dinv, buf0, n);

  // edge phase: gather + scatter-add (L2-resident tables)
  const int tot = e + n;
  k_scatter<<<(tot + 7) / 8, 256, 0, stream>>>(row, col, buf0, dinv, buf3, e, n);

  // update GEMM + classifier fused, writes d_out directly
  k_gemm_upd_cls<<<mt, 256, 0, stream>>>(buf3, dinv, W_upd, b_upd, W_cls, b_cls,
                                         (float*)d_out, n);
}